// Decoder_17179869184649
// MI455X (gfx1250) — compile-verified
//
#include <hip/hip_runtime.h>
#include <hip/hip_bf16.h>
#include <math.h>

#define HIDDEN 1024
#define VOCAB  50257
#define KHALF  512   // K-split: each wave handles half of the K=1024 reduction

typedef __attribute__((ext_vector_type(2))) float v2f;
typedef __attribute__((ext_vector_type(8))) float v8f;

__device__ __forceinline__ float sigmoid_f(float x) {
    return 1.0f / (1.0f + expf(-x));
}

// ---------------------------------------------------------------------------
// Gather: x = emb[input_id], h = h0, c = c0
// ---------------------------------------------------------------------------
__global__ void gather_kernel(const int* __restrict__ input_id,
                              const float* __restrict__ emb,
                              const float* __restrict__ h0,
                              const float* __restrict__ c0,
                              float* __restrict__ x,
                              float* __restrict__ h,
                              float* __restrict__ c) {
    int j = blockIdx.x * blockDim.x + threadIdx.x;
    if (j < HIDDEN) {
        int id = input_id[0];
        x[j] = emb[(size_t)id * HIDDEN + j];
        h[j] = h0[j];
        c[j] = c0[j];
    }
}

// ---------------------------------------------------------------------------
// WMMA GEMV: y[r] = W1[r,:]·f(v1) (+ W2[r,:]·v2) (+ b1[r]) (+ b2[r])
//   f = relu if RELU1.  K fixed = HIDDEN = 1024.
//
// Block = 256 threads = 8 waves covering 64 rows.
//   wave w: row tile t = w>>1 (16 rows), K-half kh = w&1 (512 elements).
// Per wave, V_WMMA_F32_16X16X4_F32:
//   A (16x4 f32): lanes m / m+16 hold row row_base+m; lane<16 -> K{0,1},
//                 lane>=16 -> K{2,3}  => per-lane float2, pairs of lanes
//                 fetch 16 contiguous bytes per row (coalesced stream).
//   B (4x16 f32): column 0 only (lane0: v[k],v[k+1]; lane16: v[k+2],v[k+3]).
//   D column 0 = 16-row partial, in lanes 0/16 VGPRs 0..7.
// 4 accumulators break the WMMA D->C chain; A is register double-buffered so
// loadcnt waits retire loads issued a full iteration earlier; global_prefetch
// runs ~2KB ahead so demand loads hit L2. K-half partials combine via LDS.
// ---------------------------------------------------------------------------
template <bool HAS2, bool RELU1>
__global__ __launch_bounds__(256) void wmma_gemv_kernel(
    const float* __restrict__ W1, const float* __restrict__ v1,
    const float* __restrict__ W2, const float* __restrict__ v2,
    const float* __restrict__ b1, const float* __restrict__ b2,
    float* __restrict__ y, int n_rows) {
    __shared__ float s1[HIDDEN];
    __shared__ float s2[HIDDEN];
    __shared__ float part[8][16];

    const int tid = threadIdx.x;
    for (int i = tid; i < HIDDEN; i += 256) {
        float xv = v1[i];
        s1[i] = RELU1 ? fmaxf(xv, 0.0f) : xv;
        if (HAS2) s2[i] = v2[i];
    }
    __syncthreads();

    const int wave = tid >> 5;
    const int lane = tid & 31;
    const int m    = lane & 15;
    const int koff = (lane >> 4) << 1;        // 0 or 2
    const int tile = wave >> 1;               // 0..3 (16-row tile in block)
    const int kh   = wave & 1;                // K-half
    const int kbase = kh * KHALF;

    const int block_base = blockIdx.x * 64;
    const int row_base   = block_base + tile * 16;

    int row = row_base + m;
    if (row >= n_rows) row = n_rows - 1;      // clamp loads on edge tiles

    v8f acc0 = {}; v8f acc1 = {}; v8f acc2 = {}; v8f acc3 = {};
    const v2f zz = {0.0f, 0.0f};
    const bool col0 = (m == 0);

    {
        const float* wp = W1 + (size_t)row * HIDDEN + kbase + koff;
        const float* sp = s1 + kbase + koff;
        v2f ca0 = *(const v2f*)(wp);
        v2f ca1 = *(const v2f*)(wp + 4);
        v2f ca2 = *(const v2f*)(wp + 8);
        v2f ca3 = *(const v2f*)(wp + 12);
        for (int k = 0; k < KHALF; k += 16) {
            const int kn = (k + 16 < KHALF) ? (k + 16) : k;
            v2f na0 = *(const v2f*)(wp + kn);
            v2f na1 = *(const v2f*)(wp + kn + 4);
            v2f na2 = *(const v2f*)(wp + kn + 8);
            v2f na3 = *(const v2f*)(wp + kn + 12);
            __builtin_prefetch(wp + k + 512, 0, 1);
            v2f t0 = *(const v2f*)(sp + k);
            v2f t1 = *(const v2f*)(sp + k + 4);
            v2f t2 = *(const v2f*)(sp + k + 8);
            v2f t3 = *(const v2f*)(sp + k + 12);
            v2f b0v = col0 ? t0 : zz;
            v2f b1v = col0 ? t1 : zz;
            v2f b2v = col0 ? t2 : zz;
            v2f b3v = col0 ? t3 : zz;
            acc0 = __builtin_amdgcn_wmma_f32_16x16x4_f32(false, ca0, false, b0v, (short)0, acc0, false, false);
            acc1 = __builtin_amdgcn_wmma_f32_16x16x4_f32(false, ca1, false, b1v, (short)0, acc1, false, false);
            acc2 = __builtin_amdgcn_wmma_f32_16x16x4_f32(false, ca2, false, b2v, (short)0, acc2, false, false);
            acc3 = __builtin_amdgcn_wmma_f32_16x16x4_f32(false, ca3, false, b3v, (short)0, acc3, false, false);
            ca0 = na0; ca1 = na1; ca2 = na2; ca3 = na3;
        }
    }

    if (HAS2) {
        const float* wp = W2 + (size_t)row * HIDDEN + kbase + koff;
        const float* sp = s2 + kbase + koff;
        v2f ca0 = *(const v2f*)(wp);
        v2f ca1 = *(const v2f*)(wp + 4);
        v2f ca2 = *(const v2f*)(wp + 8);
        v2f ca3 = *(const v2f*)(wp + 12);
        for (int k = 0; k < KHALF; k += 16) {
            const int kn = (k + 16 < KHALF) ? (k + 16) : k;
            v2f na0 = *(const v2f*)(wp + kn);
            v2f na1 = *(const v2f*)(wp + kn + 4);
            v2f na2 = *(const v2f*)(wp + kn + 8);
            v2f na3 = *(const v2f*)(wp + kn + 12);
            __builtin_prefetch(wp + k + 512, 0, 1);
            v2f t0 = *(const v2f*)(sp + k);
            v2f t1 = *(const v2f*)(sp + k + 4);
            v2f t2 = *(const v2f*)(sp + k + 8);
            v2f t3 = *(const v2f*)(sp + k + 12);
            v2f b0v = col0 ? t0 : zz;
            v2f b1v = col0 ? t1 : zz;
            v2f b2v = col0 ? t2 : zz;
            v2f b3v = col0 ? t3 : zz;
            acc0 = __builtin_amdgcn_wmma_f32_16x16x4_f32(false, ca0, false, b0v, (short)0, acc0, false, false);
            acc1 = __builtin_amdgcn_wmma_f32_16x16x4_f32(false, ca1, false, b1v, (short)0, acc1, false, false);
            acc2 = __builtin_amdgcn_wmma_f32_16x16x4_f32(false, ca2, false, b2v, (short)0, acc2, false, false);
            acc3 = __builtin_amdgcn_wmma_f32_16x16x4_f32(false, ca3, false, b3v, (short)0, acc3, false, false);
            ca0 = na0; ca1 = na1; ca2 = na2; ca3 = na3;
        }
    }

    v8f acc = acc0 + acc1 + acc2 + acc3;

    // Partial D column 0: lane0 vgpr g -> local row g, lane16 -> local row 8+g
    if (col0) {
        const int rb = (lane >> 4) << 3;
#pragma unroll
        for (int g = 0; g < 8; ++g) part[wave][rb + g] = acc[g];
    }
    __syncthreads();

    // Combine the two K-half partials per row; add biases; store.
    if (tid < 64) {
        const int r = block_base + tid;
        if (r < n_rows) {
            const int t   = tid >> 4;
            const int idx = tid & 15;
            float val = part[2 * t][idx] + part[2 * t + 1][idx];
            if (b1) val += b1[r];
            if (b2) val += b2[r];
            y[r] = val;
        }
    }
}

// ---------------------------------------------------------------------------
// LSTM cell update (torch gate order i,f,g,o along dim0 of gates[4096])
// ---------------------------------------------------------------------------
__global__ void lstm_update_kernel(const float* __restrict__ gates,
                                   float* __restrict__ c_buf,
                                   float* __restrict__ h_buf) {
    int j = blockIdx.x * blockDim.x + threadIdx.x;
    if (j < HIDDEN) {
        float ig = gates[j];
        float fg = gates[HIDDEN + j];
        float gg = gates[2 * HIDDEN + j];
        float og = gates[3 * HIDDEN + j];
        float c  = sigmoid_f(fg) * c_buf[j] + sigmoid_f(ig) * tanhf(gg);
        float h  = sigmoid_f(og) * tanhf(c);
        c_buf[j] = c;
        h_buf[j] = h;
    }
}

// ---------------------------------------------------------------------------
// Single-block max + log-sum-exp over logits
// ---------------------------------------------------------------------------
__global__ __launch_bounds__(1024) void logsumexp_kernel(
    const float* __restrict__ logits, float* __restrict__ logZ, int n) {
    __shared__ float red[1024];
    int tid = threadIdx.x;

    float mx = -INFINITY;
    for (int i = tid; i < n; i += 1024) mx = fmaxf(mx, logits[i]);
    red[tid] = mx;
    __syncthreads();
    for (int s = 512; s > 0; s >>= 1) {
        if (tid < s) red[tid] = fmaxf(red[tid], red[tid + s]);
        __syncthreads();
    }
    float mval = red[0];
    __syncthreads();

    float sum = 0.0f;
    for (int i = tid; i < n; i += 1024) sum += expf(logits[i] - mval);
    red[tid] = sum;
    __syncthreads();
    for (int s = 512; s > 0; s >>= 1) {
        if (tid < s) red[tid] += red[tid + s];
        __syncthreads();
    }
    if (tid == 0) logZ[0] = mval + logf(red[0]);
}

// ---------------------------------------------------------------------------
// out[0:V) = logits - logZ ; out[V:V+H) = h ; out[V+H:V+2H) = c
// ---------------------------------------------------------------------------
__global__ void finalize_kernel(const float* __restrict__ logits,
                                const float* __restrict__ logZ,
                                const float* __restrict__ h,
                                const float* __restrict__ c,
                                float* __restrict__ out, int n) {
    int i = blockIdx.x * blockDim.x + threadIdx.x;
    if (i < n) out[i] = logits[i] - logZ[0];
    if (i < HIDDEN) {
        out[n + i] = h[i];
        out[n + HIDDEN + i] = c[i];
    }
}

// ---------------------------------------------------------------------------
extern "C" void kernel_launch(void* const* d_in, const int* in_sizes, int n_in,
                              void* d_out, int out_size, void* d_ws, size_t ws_size,
                              hipStream_t stream) {
    const int*   input_id = (const int*)d_in[0];
    const float* h0    = (const float*)d_in[1];
    const float* c0    = (const float*)d_in[2];
    const float* emb   = (const float*)d_in[3];
    const float* W_ih  = (const float*)d_in[4];
    const float* W_hh  = (const float*)d_in[5];
    const float* b_ih  = (const float*)d_in[6];
    const float* b_hh  = (const float*)d_in[7];
    const float* W_out = (const float*)d_in[8];
    const float* b_out = (const float*)d_in[9];
    float* out = (float*)d_out;

    float* ws     = (float*)d_ws;
    float* x_buf  = ws;                  // 1024
    float* h_buf  = ws + 1024;           // 1024
    float* c_buf  = ws + 2048;           // 1024
    float* gates  = ws + 3072;           // 4096
    float* logits = ws + 7168;           // 50304 (padded)
    float* logZ   = ws + 7168 + 50304;   // 1

    gather_kernel<<<4, 256, 0, stream>>>(input_id, emb, h0, c0, x_buf, h_buf, c_buf);

    const float* xin = x_buf;
    for (int layer = 0; layer < 2; ++layer) {
        // gates = W_ih·relu(x) + b_ih + W_hh·h + b_hh  (4096 rows -> 64 blocks)
        wmma_gemv_kernel<true, true><<<(4 * HIDDEN) / 64, 256, 0, stream>>>(
            W_ih, xin, W_hh, h_buf, b_ih, b_hh, gates, 4 * HIDDEN);
        lstm_update_kernel<<<4, 256, 0, stream>>>(gates, c_buf, h_buf);
        xin = h_buf;  // x for layer 2 is h1 (relu applied inside gemv)
    }

    // logits = W_out·h + b_out  (50257 rows -> 786 blocks of 8 waves)
    int vblocks = (VOCAB + 63) / 64;
    wmma_gemv_kernel<false, false><<<vblocks, 256, 0, stream>>>(
        W_out, h_buf, nullptr, nullptr, b_out, nullptr, logits, VOCAB);

    logsumexp_kernel<<<1, 1024, 0, stream>>>(logits, logZ, VOCAB);

    finalize_kernel<<<(VOCAB + 255) / 256, 256, 0, stream>>>(
        logits, logZ, h_buf, c_buf, out, VOCAB);
}